// MultiCrossEntropyLoss_55989193670919
// MI455X (gfx1250) — compile-verified
//
#include <hip/hip_runtime.h>
#include <hip/hip_bf16.h>

// MultiCrossEntropyLoss: loss = -(1/N) * sum_i log(predicts[i, labels[i]])
// N = 1048576, C = 256.  Gather-bound: ~140 MB effective HBM traffic (one
// 128B line per picked float + labels) -> ~6us floor at 23.3 TB/s.
// V_WMMA_F32_16X16X4_F32 with B=ones performs the in-wave cross-lane
// reduction on the CDNA5 matrix path.

typedef __attribute__((ext_vector_type(2))) float v2f;
typedef __attribute__((ext_vector_type(8))) float v8f;

#define CE_TPB   256
#define CE_WAVES (CE_TPB / 32)
#define CE_EPT   2            // elements per thread in the gather kernel

#define LN2_F 0.69314718055994530942f

// Cross-lane wave32 reduction via D = A x ones + 0.
// A (16x4 f32): lane m (0-15) vgpr0 = A[m][0]; lane m+16 vgpr0 = A[m][2].
// With a = {s, 0}: D[m][n] = s_m + s_{m+16} for every column n.
// Lanes 0-15 hold rowsums 0..7 in c[0..7]; lanes 16-31 hold rowsums 8..15.
// => laneSum(lane 0) + laneSum(lane 16) == sum of s over all 32 lanes.
__device__ __forceinline__ float wave_wmma_partial(float s) {
  v2f a; a.x = s;    a.y = 0.0f;
  v2f b; b.x = 1.0f; b.y = 1.0f;   // B = all ones (4x16)
  v8f c = {};
  c = __builtin_amdgcn_wmma_f32_16x16x4_f32(
      /*neg_a=*/false, a, /*neg_b=*/false, b,
      /*c_mod=*/(short)0, c, /*reuse_a=*/false, /*reuse_b=*/false);
  float r = 0.0f;
#pragma unroll
  for (int i = 0; i < 8; ++i) r += c[i];
  return r;
}

// Per-wave WMMA partials -> LDS -> total (valid on thread 0 of the block).
__device__ __forceinline__ float block_reduce_wmma(float s, float* lds) {
  const int wave = threadIdx.x >> 5;
  const int lane = threadIdx.x & 31;
  const float r = wave_wmma_partial(s);       // uniform control flow here
  if (lane == 0)       lds[wave * 2 + 0] = r;
  else if (lane == 16) lds[wave * 2 + 1] = r;
  __syncthreads();
  float t = 0.0f;
  if (threadIdx.x == 0) {
#pragma unroll
    for (int i = 0; i < CE_WAVES * 2; ++i) t += lds[i];
  }
  return t;
}

__global__ __launch_bounds__(CE_TPB)
void ce_gather_partials(const float* __restrict__ predicts,
                        const int*   __restrict__ labels,
                        float*       __restrict__ partials,
                        int n, int c) {
  __shared__ float lds[CE_WAVES * 2];
  const int tid = blockIdx.x * CE_TPB + threadIdx.x;
  const long long base = (long long)tid * CE_EPT;

  float s = 0.0f;
  if (base + 1 < n) {
    // One coalesced b64 load for both labels of this thread.
    const int2 lbl = *(const int2*)(labels + base);
    const float* r0 = predicts + (size_t)(base + 0) * (size_t)c + (size_t)lbl.x;
    const float* r1 = predicts + (size_t)(base + 1) * (size_t)c + (size_t)lbl.y;
    // Each cache line is touched exactly once -> non-temporal gather.
    const float p0 = __builtin_nontemporal_load(r0);
    const float p1 = __builtin_nontemporal_load(r1);
    // Inputs are in [1e-6, 1): raw v_log_f32 (log2) * ln2, no denorm rescue.
    s = (__builtin_amdgcn_logf(p0) + __builtin_amdgcn_logf(p1)) * LN2_F;
  } else if (base < n) {            // tail (never taken for N=2^20)
    const int   lbl = labels[base];
    const float p   = __builtin_nontemporal_load(
                        predicts + (size_t)base * (size_t)c + (size_t)lbl);
    s = __builtin_amdgcn_logf(p) * LN2_F;
  }

  const float t = block_reduce_wmma(s, lds);
  if (threadIdx.x == 0) partials[blockIdx.x] = t;
}

__global__ __launch_bounds__(CE_TPB)
void ce_finalize(const float* __restrict__ partials,
                 int m, float inv_n,
                 float* __restrict__ out) {
  __shared__ float lds[CE_WAVES * 2];
  float s = 0.0f;
  for (int i = threadIdx.x; i < m; i += CE_TPB) s += partials[i];

  const float t = block_reduce_wmma(s, lds);
  if (threadIdx.x == 0) out[0] = -t * inv_n;
}

extern "C" void kernel_launch(void* const* d_in, const int* in_sizes, int n_in,
                              void* d_out, int out_size, void* d_ws, size_t ws_size,
                              hipStream_t stream) {
  const float* predicts = (const float*)d_in[0];
  const int*   labels   = (const int*)d_in[1];   // harness: integer -> const int*
  float*       out      = (float*)d_out;
  float*       partials = (float*)d_ws;          // blocks * 4 bytes (8 KB)

  const int n = in_sizes[1];                     // N = number of rows/labels
  const int c = in_sizes[0] / (n > 0 ? n : 1);   // C = 256

  const int blocks = (n + CE_TPB * CE_EPT - 1) / (CE_TPB * CE_EPT);  // 2048

  ce_gather_partials<<<blocks, CE_TPB, 0, stream>>>(predicts, labels, partials, n, c);
  ce_finalize<<<1, CE_TPB, 0, stream>>>(partials, blocks, 1.0f / (float)n, out);
}